// SpikeFP32Linear_MultiPrecision_22351009808883
// MI455X (gfx1250) — compile-verified
//
#include <hip/hip_runtime.h>

#define B_DIM   256
#define IN_DIM  1024
#define OUT_DIM 1024

typedef __attribute__((ext_vector_type(2))) float v2f;
typedef __attribute__((ext_vector_type(8))) float v8f;

// ---------------------------------------------------------------------------
// Pass 1: pulse-bits [N,32] (floats 0/1, bit k = IEEE bit k) -> packed fp32 [N]
// Pure streaming: 128B read per element -> 4B write. Coalesced float4 loads.
// ---------------------------------------------------------------------------
__global__ void __launch_bounds__(256)
decode_pulse_kernel(const float* __restrict__ pulse, float* __restrict__ dst, int n) {
  int e = blockIdx.x * blockDim.x + threadIdx.x;
  if (e >= n) return;
  const float4* p = (const float4*)(pulse + (size_t)e * 32);
  unsigned u = 0u;
#pragma unroll
  for (int q = 0; q < 8; ++q) {
    float4 v = p[q];
    u |= (v.x != 0.0f ? 1u : 0u) << (4 * q + 0);
    u |= (v.y != 0.0f ? 1u : 0u) << (4 * q + 1);
    u |= (v.z != 0.0f ? 1u : 0u) << (4 * q + 2);
    u |= (v.w != 0.0f ? 1u : 0u) << (4 * q + 3);
  }
  dst[e] = __uint_as_float(u);
}

// ---------------------------------------------------------------------------
// Pass 2: fp32 GEMM via V_WMMA_F32_16X16X4_F32, fused bit-encode epilogue.
// One wave -> one 16x16 tile of Y. 4 waves/block -> 16 batch rows x 64 outs.
//
// A (16x4, M=rows of x_f):  lane<16: m=lane,   holds K = 0,1 of the step
//                           lane>=16: m=lane-16, holds K = 2,3
// B (4x16, N=rows of w_f):  lane<16: n=lane,   holds K = 0,1
//                           lane>=16: n=lane-16, holds K = 2,3
// C/D (16x16 f32, 8 VGPRs): VGPR r = row (lane>=16 ? 8 : 0)+r, col lane&15.
// ---------------------------------------------------------------------------
__global__ void __launch_bounds__(128)
wmma_gemm_encode_kernel(const float* __restrict__ xf, const float* __restrict__ wf,
                        float* __restrict__ out) {
  const int lane = threadIdx.x & 31;
  const int wave = threadIdx.x >> 5;              // 0..3
  const int b0   = blockIdx.x * 16;               // batch tile origin
  const int o0   = blockIdx.y * 64 + wave * 16;   // output tile origin

  const int half = lane >> 4;                     // 0 or 1
  const int mn   = lane & 15;

  // Per-lane streaming pointers: contiguous 8B per K-step, rows stay in L2.
  const float* xrow = xf + (size_t)(b0 + mn) * IN_DIM + half * 2;
  const float* wrow = wf + (size_t)(o0 + mn) * IN_DIM + half * 2;

  v8f c = {};
#pragma unroll 8
  for (int i0 = 0; i0 < IN_DIM; i0 += 4) {
    v2f a = *(const v2f*)(xrow + i0);
    v2f b = *(const v2f*)(wrow + i0);
    // 8 args: (neg_a, A, neg_b, B, c_mod, C, reuse_a, reuse_b)
    c = __builtin_amdgcn_wmma_f32_16x16x4_f32(
        /*neg_a=*/false, a, /*neg_b=*/false, b,
        /*c_mod=*/(short)0, c, /*reuse_a=*/false, /*reuse_b=*/false);
  }

  // Encode epilogue: expand each fp32 result into 32 pulse floats
  // (8 aligned float4 stores per value, 128B contiguous per output element).
#pragma unroll
  for (int r = 0; r < 8; ++r) {
    unsigned u  = __float_as_uint(c[r]);
    int brow    = b0 + half * 8 + r;
    int ocol    = o0 + mn;
    float4* dst = (float4*)(out + ((size_t)brow * OUT_DIM + ocol) * 32);
#pragma unroll
    for (int q = 0; q < 8; ++q) {
      float4 f;
      f.x = (float)((u >> (4 * q + 0)) & 1u);
      f.y = (float)((u >> (4 * q + 1)) & 1u);
      f.z = (float)((u >> (4 * q + 2)) & 1u);
      f.w = (float)((u >> (4 * q + 3)) & 1u);
      dst[q] = f;
    }
  }
}

extern "C" void kernel_launch(void* const* d_in, const int* in_sizes, int n_in,
                              void* d_out, int out_size, void* d_ws, size_t ws_size,
                              hipStream_t stream) {
  const float* x_pulse = (const float*)d_in[0];   // [B, IN, 32]
  const float* w_pulse = (const float*)d_in[1];   // [OUT, IN, 32]
  float*       out     = (float*)d_out;           // [B, OUT, 32]

  float* xf = (float*)d_ws;                       // [B, IN]   1 MB
  float* wf = xf + (size_t)B_DIM * IN_DIM;        // [OUT, IN] 4 MB

  const int nx = B_DIM * IN_DIM;
  const int nw = OUT_DIM * IN_DIM;
  decode_pulse_kernel<<<(nx + 255) / 256, 256, 0, stream>>>(x_pulse, xf, nx);
  decode_pulse_kernel<<<(nw + 255) / 256, 256, 0, stream>>>(w_pulse, wf, nw);

  dim3 grid(B_DIM / 16, OUT_DIM / 64);            // (16, 16)
  wmma_gemm_encode_kernel<<<grid, 128, 0, stream>>>(xf, wf, out);
}